// LengthRegulatorWithDurations_59562606461252
// MI455X (gfx1250) — compile-verified
//
#include <hip/hip_runtime.h>
#include <cstdint>

namespace {

constexpr int kB    = 16;
constexpr int kT    = 512;
constexpr int kD    = 384;
constexpr int kVec  = kD / 4;    // 96 float4 per output row
constexpr int kRows = 128;       // output rows per block (chunk)

typedef float v4f __attribute__((ext_vector_type(4)));

// ---------------------------------------------------------------------------
// Kernel A: per-batch inclusive scan of durations.
//   cum_ws[b][t] = sum(dur[b][0..t]),  mel_lens_out[b] = cum[b][T-1] (float)
// One 512-thread block (16 wave32) per batch; Hillis-Steele scan in LDS.
// ---------------------------------------------------------------------------
__global__ __launch_bounds__(kT)
void duration_scan_kernel(const int* __restrict__ durations,
                          int* __restrict__ cum_ws,
                          float* __restrict__ mel_lens_out) {
  __shared__ int s[kT];
  const int b = blockIdx.x;
  const int t = threadIdx.x;

  s[t] = durations[b * kT + t];
  __syncthreads();

#pragma unroll
  for (int off = 1; off < kT; off <<= 1) {
    const int v = (t >= off) ? s[t - off] : 0;
    __syncthreads();
    if (t >= off) s[t] += v;
    __syncthreads();
  }

  cum_ws[b * kT + t] = s[t];
  if (t == kT - 1) {
    // reference returns un-clipped cum[:, -1]; output buffer dtype is float
    mel_lens_out[b] = (float)s[t];
  }
}

// ---------------------------------------------------------------------------
// Kernel B: expand.  grid = (ML/kRows, B), 256 threads/block (8 wave32).
// Phase 0: CDNA5 async global->LDS pull of the batch's 2KB cumsum table
//          (ASYNCcnt + s_wait_asynccnt + workgroup barrier).
// Phase 1: one searchsorted per output row of this 128-row chunk -> s_tok.
// Phase 2: stream 128x96 float4: 1 ds_load (token id) + 1 b128 gather +
//          1 b128 NON-TEMPORAL store (output is write-once; keep L2 for
//          the encoder rows which are re-read ~3.5x each).
// ---------------------------------------------------------------------------
__global__ __launch_bounds__(256)
void expand_kernel(const float* __restrict__ enc,
                   const int* __restrict__ cum_ws,
                   float* __restrict__ out,
                   int max_len) {
  __shared__ int s_cum[kT];
  __shared__ int s_tok[kRows];

  const int b        = blockIdx.y;
  const int tid      = threadIdx.x;
  const int row_base = blockIdx.x * kRows;

  // --- Phase 0: async copy cum[b][0..511] (2048 B): 256 lanes x b64 ---
  {
    const unsigned lds_off =
        (unsigned)(uintptr_t)(&s_cum[0]) + (unsigned)(tid * 8);
    const unsigned voff  = (unsigned)(b * kT * 4 + tid * 8);
    const uint64_t sbase = (uint64_t)(uintptr_t)cum_ws;
    asm volatile("global_load_async_to_lds_b64 %0, %1, %2"
                 :
                 : "v"(lds_off), "v"(voff), "s"(sbase)
                 : "memory");
    asm volatile("s_wait_asynccnt 0" ::: "memory");
  }
  __syncthreads();  // each wave only tracks its own ASYNCcnt; share via barrier

  const int vlen = min(s_cum[kT - 1], max_len);

  // --- Phase 1: one binary search per row of this chunk ---
  if (tid < kRows) {
    const int row = row_base + tid;
    int tok = -1;  // sentinel: emit zeros
    if (row < vlen) {
      // searchsorted(cum, row, side='right'): smallest i with cum[i] > row
      int lo = 0, hi = kT;
#pragma unroll
      for (int step = 0; step < 10; ++step) {
        const int  mid = (lo + hi) >> 1;
        const bool le  = (s_cum[mid] <= row);
        lo = le ? (mid + 1) : lo;
        hi = le ? hi : mid;
      }
      tok = (lo < kT) ? lo : (kT - 1);
    }
    s_tok[tid] = tok;
  }
  __syncthreads();

  // --- Phase 2: copy/zero 128 rows x 96 float4 ---
  const v4f* __restrict__ encv =
      (const v4f*)enc + (size_t)b * kT * kVec;
  v4f* __restrict__ outv =
      (v4f*)out + ((size_t)b * max_len + row_base) * kVec;

  constexpr int kChunkVec = kRows * kVec;  // 12288 float4
#pragma unroll 4
  for (int idx = tid; idx < kChunkVec; idx += 256) {
    const int r   = idx / kVec;
    const int c   = idx - r * kVec;
    const int row = row_base + r;
    if (row >= max_len) break;             // only for ML not multiple of 128

    const int tok = s_tok[r];
    v4f v = {0.f, 0.f, 0.f, 0.f};
    if (tok >= 0) {
      v = encv[(size_t)tok * kVec + c];    // b128 gather (RT: rows reused)
    }
    __builtin_nontemporal_store(v, &outv[idx]);  // b128 store th:NT
  }
}

} // namespace

extern "C" void kernel_launch(void* const* d_in, const int* in_sizes, int n_in,
                              void* d_out, int out_size, void* d_ws,
                              size_t ws_size, hipStream_t stream) {
  (void)in_sizes; (void)n_in; (void)ws_size;

  const float* enc = (const float*)d_in[0];
  const int*   dur = (const int*)d_in[1];
  // d_in[2] is max_len as a device scalar; recover it host-side from the
  // output layout (out_size = B*ML*D + B) to stay graph-capture safe.
  const int ML = (out_size - kB) / (kB * kD);

  float* out     = (float*)d_out;
  float* mel_out = out + (size_t)kB * ML * kD;   // tuple tail: mel_lens
  int*   cum_ws  = (int*)d_ws;                   // B*T ints = 32 KB

  duration_scan_kernel<<<dim3(kB), dim3(kT), 0, stream>>>(dur, cum_ws, mel_out);

  const int gx = (ML + kRows - 1) / kRows;       // 32 chunks per batch
  expand_kernel<<<dim3(gx, kB), dim3(256), 0, stream>>>(enc, cum_ws, out, ML);
}